// TestQKNormRopeModelWithBias_76587856822536
// MI455X (gfx1250) — compile-verified
//
#include <hip/hip_runtime.h>
#include <cstdint>

// Problem constants (from reference)
#define HEAD_DIM   128
#define NUM_HEADS  32
#define NUM_KV     8
#define SEQ_LEN    8192
#define Q_SIZE     (NUM_HEADS * HEAD_DIM)      // 4096
#define KV_SIZE    (NUM_KV * HEAD_DIM)         // 1024
#define ROW_FLOATS (Q_SIZE + 2 * KV_SIZE)      // 6144
#define EPSF       1e-6f

typedef float v4f __attribute__((ext_vector_type(4)));

// Normalize + bias + RoPE for one head-slice held by this lane.
// x: 4 contiguous elements (dims 4*lane..4*lane+3); partner lane = lane^16 holds d +/- 64.
static __device__ __forceinline__ v4f
norm_rope(v4f x, v4f w, v4f b, v4f cw, v4f sw, float sgn)
{
    float ss = x.x * x.x + x.y * x.y + x.z * x.z + x.w * x.w;
    ss += __shfl_xor(ss, 1, 32);
    ss += __shfl_xor(ss, 2, 32);
    ss += __shfl_xor(ss, 4, 32);
    ss += __shfl_xor(ss, 8, 32);
    ss += __shfl_xor(ss, 16, 32);
    const float rstd = rsqrtf(ss * (1.0f / HEAD_DIM) + EPSF);

    v4f y;
    y.x = x.x * rstd * w.x + b.x;
    y.y = x.y * rstd * w.y + b.y;
    y.z = x.z * rstd * w.z + b.z;
    y.w = x.w * rstd * w.w + b.w;

    // rotate_half: element d +/- 64 lives in lane (lane^16) at the same sub-index.
    const float px = __shfl_xor(y.x, 16, 32);
    const float py = __shfl_xor(y.y, 16, 32);
    const float pz = __shfl_xor(y.z, 16, 32);
    const float pw = __shfl_xor(y.w, 16, 32);

    v4f o;
    o.x = y.x * cw.x + sgn * px * sw.x;
    o.y = y.y * cw.y + sgn * py * sw.y;
    o.z = y.z * cw.z + sgn * pz * sw.z;
    o.w = y.w * cw.w + sgn * pw * sw.w;
    return o;
}

// One block per sequence row. 256 threads = 8 wave32s.
// Each wave handles 4 Q heads + 1 K head; one head = 32 lanes x 4 floats.
// V segment (1024 floats) streams through the CDNA5 async global<->LDS path,
// issued up-front so the Q/K compute hides its latency.
__global__ __launch_bounds__(256) void
qknorm_rope_kernel(const float* __restrict__ qkv,
                   const float* __restrict__ qw_p, const float* __restrict__ kw_p,
                   const float* __restrict__ qb_p, const float* __restrict__ kb_p,
                   const float* __restrict__ cos_p, const float* __restrict__ sin_p,
                   float* __restrict__ out)
{
    const int s    = blockIdx.x;
    const int tid  = threadIdx.x;
    const int lane = tid & 31;
    const int wave = tid >> 5;

    float* out_q = out;                               // (S, 32, 128)
    float* out_k = out + (size_t)SEQ_LEN * Q_SIZE;    // (S,  8, 128)
    float* out_v = out_k + (size_t)SEQ_LEN * KV_SIZE; // (S, 1024)

    // ---- Issue the async V copy (global -> LDS) first: latency hidden by Q/K work.
    __shared__ __align__(16) float vbuf[KV_SIZE];     // 4 KB: 256 threads x 16 B
    const unsigned lds_off =
        (unsigned)(uintptr_t)(&vbuf[tid * 4]);        // low 32 bits of flat-shared addr == LDS offset
    const unsigned long long vsrc =
        (unsigned long long)(uintptr_t)(qkv + (size_t)s * ROW_FLOATS + Q_SIZE + KV_SIZE + tid * 4);
    const unsigned long long vdst =
        (unsigned long long)(uintptr_t)(out_v + (size_t)s * KV_SIZE + tid * 4);

    asm volatile("global_load_async_to_lds_b128 %0, %1, off th:TH_LOAD_NT"
                 :: "v"(lds_off), "v"(vsrc) : "memory");

    // ---- Parameters: reused by every block -> regular-temporal loads (stay in L2).
    const v4f cw  = ((const v4f*)cos_p)[lane];
    const v4f sw  = ((const v4f*)sin_p)[lane];
    const v4f qw4 = ((const v4f*)qw_p)[lane];
    const v4f qb4 = ((const v4f*)qb_p)[lane];
    const v4f kw4 = ((const v4f*)kw_p)[lane];
    const v4f kb4 = ((const v4f*)kb_p)[lane];

    const v4f* row = (const v4f*)(qkv + (size_t)s * ROW_FLOATS);
    const float sgn = (lane < 16) ? -1.0f : 1.0f;   // rotate_half: -x2 for d<64, +x1 for d>=64

    // ---- Q heads: h = wave + 8*j, j = 0..3 (h < 32 always). Streamed data -> NT.
#pragma unroll
    for (int j = 0; j < 4; ++j) {
        const int h = wave + 8 * j;
        v4f x = __builtin_nontemporal_load(&row[h * (HEAD_DIM / 4) + lane]);
        v4f o = norm_rope(x, qw4, qb4, cw, sw, sgn);
        __builtin_nontemporal_store(
            o, (v4f*)(out_q + (size_t)s * Q_SIZE + (size_t)h * HEAD_DIM) + lane);
    }

    // ---- K head: h = wave + 32 (always a K head; kv index = wave).
    {
        const int h = wave + NUM_HEADS;
        v4f x = __builtin_nontemporal_load(&row[h * (HEAD_DIM / 4) + lane]);
        v4f o = norm_rope(x, kw4, kb4, cw, sw, sgn);
        __builtin_nontemporal_store(
            o, (v4f*)(out_k + (size_t)s * KV_SIZE + (size_t)wave * HEAD_DIM) + lane);
    }

    // ---- Drain the async V load, then push LDS -> global (write-rinse NT).
    asm volatile("s_wait_asynccnt 0" ::: "memory");
    asm volatile("global_store_async_from_lds_b128 %0, %1, off th:TH_STORE_NT"
                 :: "v"(vdst), "v"(lds_off) : "memory");
    asm volatile("s_wait_asynccnt 0" ::: "memory");
}

extern "C" void kernel_launch(void* const* d_in, const int* in_sizes, int n_in,
                              void* d_out, int out_size, void* d_ws, size_t ws_size,
                              hipStream_t stream) {
    const float* qkv  = (const float*)d_in[0];
    const float* qw   = (const float*)d_in[1];
    const float* kw   = (const float*)d_in[2];
    const float* qb   = (const float*)d_in[3];
    const float* kb   = (const float*)d_in[4];
    const float* cosw = (const float*)d_in[5];
    const float* sinw = (const float*)d_in[6];
    float* out = (float*)d_out;

    qknorm_rope_kernel<<<SEQ_LEN, 256, 0, stream>>>(qkv, qw, kw, qb, kb, cosw, sinw, out);
}